// InverseModule_7172595385018
// MI455X (gfx1250) — compile-verified
//
#include <hip/hip_runtime.h>

// Batched 4x4 float32 matrix inverse (blockwise Schur, matching the reference).
// N = 4,000,000 matrices; 512 MB total traffic @ 23.3 TB/s => ~22 us floor,
// AI ~0.8 FLOP/B => strictly HBM-bound. WMMA cannot help (independent 4x4
// inverses waste >75% of any 16x16 tile and add traffic). CDNA5 paths used:
//   - global_load_async_to_lds_b128 (ASYNCcnt) staging: 512 B contiguous per
//     wave-instruction into LDS, decoupled from VGPRs.
//   - XOR-swizzled LDS layout: conflict-free ds_load_b128 tile reads
//     (unswizzled layout has 4-way bank conflicts at 64-byte lane stride).
//   - Non-temporal load/store hints: data is touched exactly once, keep it
//     out of the 192 MB L2's working set.

#define BLOCK 256                         // 8 wave32 waves per workgroup
#define CHUNKS_PER_MAT 4                  // 4 x 16 B chunks = one 4x4 matrix

typedef float v4f __attribute__((ext_vector_type(4)));

struct M2 { float a, b, c, d; };          // row-major 2x2: [a b; c d]

__device__ __forceinline__ M2 mul2(M2 x, M2 y) {
    return { x.a * y.a + x.b * y.c, x.a * y.b + x.b * y.d,
             x.c * y.a + x.d * y.c, x.c * y.b + x.d * y.d };
}
__device__ __forceinline__ M2 sub2(M2 x, M2 y) {
    return { x.a - y.a, x.b - y.b, x.c - y.c, x.d - y.d };
}
__device__ __forceinline__ M2 neg2(M2 x) {
    return { -x.a, -x.b, -x.c, -x.d };
}
__device__ __forceinline__ M2 inv2(M2 m) {
    float det = m.a * m.d - m.b * m.c;
    float r = 1.0f / det;
    return { m.d * r, -m.b * r, -m.c * r, m.a * r };
}

// 16B-chunk XOR swizzle: folds the 256B-row index (bits [5:4]) into the
// chunk-column (bits [1:0]). Involution; makes both the async writes and the
// strided ds_load_b128 reads hit all 16 chunk-columns per half-wave.
__device__ __forceinline__ unsigned swz(unsigned c) {
    return c ^ ((c >> 4) & 3u);
}

__global__ void __launch_bounds__(BLOCK)
inv4x4_kernel(const float* __restrict__ x, float* __restrict__ out, int n) {
    __shared__ __align__(16) v4f lds[BLOCK * CHUNKS_PER_MAT];   // 16 KB

    const int tid = threadIdx.x;
    const int blockStart = blockIdx.x * BLOCK;
    const int idx = blockStart + tid;

    v4f r0, r1, r2, r3;

    if (blockStart + BLOCK <= n) {
        // ---- Async-stage 16 KB global -> LDS, coalesced + swizzled ----
        const char* gbase = (const char*)x + (size_t)blockStart * 64;
        // Low 32 bits of a generic shared pointer == LDS byte offset
        // (ISA 10.2: LDS aperture truncates to addr[31:0]).
        const unsigned ldsBase = (unsigned)(size_t)(&lds[0]);
#pragma unroll
        for (int k = 0; k < 4; ++k) {
            unsigned c = (unsigned)tid + (unsigned)k * BLOCK;  // linear chunk
            unsigned ldsOff = ldsBase + swz(c) * 16u;
            unsigned long long gaddr =
                (unsigned long long)(gbase + (size_t)c * 16u);
            asm volatile("global_load_async_to_lds_b128 %0, %1, off th:TH_LOAD_NT"
                         :: "v"(ldsOff), "v"(gaddr)
                         : "memory");
        }
        // Drain this wave's ASYNCcnt, then barrier so all waves' slices are
        // visible before the ds reads.
        asm volatile("s_wait_asynccnt 0" ::: "memory");
        __syncthreads();

        r0 = lds[swz((unsigned)tid * 4u + 0u)];
        r1 = lds[swz((unsigned)tid * 4u + 1u)];
        r2 = lds[swz((unsigned)tid * 4u + 2u)];
        r3 = lds[swz((unsigned)tid * 4u + 3u)];
    } else {
        // ---- Tail block (unused for N=4M; kept for generality) ----
        if (idx >= n) return;
        const v4f* m = (const v4f*)(x + (size_t)idx * 16);
        r0 = m[0]; r1 = m[1]; r2 = m[2]; r3 = m[3];
    }

    // ---- Blockwise Schur inverse (matches reference algorithm) ----
    M2 a { r0.x, r0.y, r1.x, r1.y };
    M2 b { r0.z, r0.w, r1.z, r1.w };
    M2 c { r2.x, r2.y, r3.x, r3.y };
    M2 d { r2.z, r2.w, r3.z, r3.w };

    M2 ai  = inv2(a);                               // a^-1
    M2 cai = mul2(c, ai);                           // c a^-1
    M2 aib = mul2(ai, b);                           // a^-1 b
    M2 s   = inv2(sub2(d, mul2(cai, b)));           // (d - c a^-1 b)^-1
    M2 B   = neg2(mul2(aib, s));                    // -(a^-1 b s)
    M2 C   = neg2(mul2(s, cai));                    // -(s c a^-1)
    M2 A   = sub2(ai, mul2(B, cai));                // a^-1 + a^-1 b s c a^-1
    M2 D   = s;

    // ---- Non-temporal b128 stores (write-once stream) ----
    v4f* o = (v4f*)(out + (size_t)idx * 16);
    v4f o0 = { A.a, A.b, B.a, B.b };
    v4f o1 = { A.c, A.d, B.c, B.d };
    v4f o2 = { C.a, C.b, D.a, D.b };
    v4f o3 = { C.c, C.d, D.c, D.d };
    __builtin_nontemporal_store(o0, o + 0);
    __builtin_nontemporal_store(o1, o + 1);
    __builtin_nontemporal_store(o2, o + 2);
    __builtin_nontemporal_store(o3, o + 3);
}

extern "C" void kernel_launch(void* const* d_in, const int* in_sizes, int n_in,
                              void* d_out, int out_size, void* d_ws, size_t ws_size,
                              hipStream_t stream) {
    const float* x = (const float*)d_in[0];
    float* out = (float*)d_out;
    const int n = in_sizes[0] / 16;                  // number of 4x4 matrices
    const int blocks = (n + BLOCK - 1) / BLOCK;      // 15625 for N = 4,000,000
    inv4x4_kernel<<<blocks, BLOCK, 0, stream>>>(x, out, n);
}